// RadarDC_12300786336443
// MI455X (gfx1250) — compile-verified
//
#include <hip/hip_runtime.h>

#define W_DIM 1600
#define B_DIM 4
#define H_DIM 900
#define NPL   29            // ceil(H/32) elements per lane
#define WPB   8             // waves (columns) per block
#define TPB   (WPB * 32)

#if __has_builtin(__builtin_amdgcn_global_store_async_from_lds_b128)
#define USE_ASYNC_STORE 1
typedef int v4i __attribute__((vector_size(16)));
typedef __attribute__((address_space(1))) v4i* gv4i_p;   // global (AS1)
typedef __attribute__((address_space(3))) v4i* lv4i_p;   // LDS (AS3)
#else
#define USE_ASYNC_STORE 0
#endif

__device__ __forceinline__ void wait_async_cnt0() {
#if __has_builtin(__builtin_amdgcn_s_wait_asynccnt)
  __builtin_amdgcn_s_wait_asynccnt(0);
#else
  asm volatile("s_wait_asynccnt 0" ::: "memory");
#endif
}

__global__ __launch_bounds__(TPB) void radar_dc_kernel(
    const float* __restrict__ radar,
    const float* __restrict__ mde,
    float* __restrict__ out)
{
  // tile[y*WPB + wi] : placement results for WPB consecutive-w columns (same b)
  __shared__ float tile[H_DIM * WPB];

  const int lane = threadIdx.x & 31;
  const int wi   = threadIdx.x >> 5;            // wave id in block == column slot
  const int w    = blockIdx.x * WPB + wi;
  const int b    = blockIdx.y;
  const int col  = w * B_DIM + b;               // input column index: (w*B+b)

  const float* __restrict__ rcol = radar + (size_t)col * H_DIM;
  const float* __restrict__ mcol = mde   + (size_t)col * H_DIM;

  // ---- setup: mde column -> registers, prefetch radar, zero own tile slice ----
  float mv[NPL];
  unsigned mvalid = 0u;                         // bit j: mde[j*32+lane] valid
#pragma unroll
  for (int j = 0; j < NPL; ++j) {
    const int y = j * 32 + lane;
    float m = 0.0f;
    if (y < H_DIM) {
      m = mcol[y];
      __builtin_prefetch(rcol + y, 0, 3);       // gfx1250 global_prefetch_b8
      tile[y * WPB + wi] = 0.0f;
      if (m != 0.0f) mvalid |= (1u << j);
    }
    mv[j] = m;
  }
  const bool has_mde = (__ballot(mvalid != 0u) != 0ull);

  // occupancy bitmask: bit j set <=> slot y=j*32+lane occupied (or out of range)
  unsigned occ = 0u;
#pragma unroll
  for (int j = 0; j < NPL; ++j)
    if (j * 32 + lane >= H_DIM) occ |= (1u << j);

  // ---- sequential placement over valid radar points, ascending y ----
#pragma unroll 1
  for (int k = 0; k < NPL; ++k) {
    const int yk = k * 32 + lane;
    const float d = (yk < H_DIM) ? rcol[yk] : 0.0f;
    unsigned mask = (unsigned)__ballot(d != 0.0f);
    while (mask) {
      const int src = __builtin_ctz(mask);
      mask &= mask - 1u;
      const float depth = __shfl(d, src, 32);
      const int y = k * 32 + src;

      // argmin over |mde - depth| with first-index tiebreak
      float bv = 1e30f;
      int   bi = 0x7fffffff;
#pragma unroll
      for (int j = 0; j < NPL; ++j) {
        const float diff = __builtin_fabsf(mv[j] - depth);
        if (((mvalid >> j) & 1u) && diff < bv) { bv = diff; bi = j * 32 + lane; }
      }
#pragma unroll
      for (int off = 16; off; off >>= 1) {
        const float v2 = __shfl_xor(bv, off, 32);
        const int   i2 = __shfl_xor(bi, off, 32);
        if (v2 < bv || (v2 == bv && i2 < bi)) { bv = v2; bi = i2; }
      }
      const int best_y = has_mde ? bi : y;

      // nearest free slot: key = 2*|dy| + (dy<0), key(best_y)=0
      int bkey = 0x7fffffff;
      int bidx = 0;
      const unsigned freeb = ~occ;
#pragma unroll
      for (int j = 0; j < NPL; ++j) {
        if ((freeb >> j) & 1u) {
          const int yy  = j * 32 + lane;
          const int dy  = yy - best_y;
          const int ad  = dy < 0 ? -dy : dy;
          const int key = 2 * ad + (dy < 0 ? 1 : 0);
          if (key < bkey) { bkey = key; bidx = yy; }
        }
      }
#pragma unroll
      for (int off = 16; off; off >>= 1) {
        const int k2 = __shfl_xor(bkey, off, 32);
        const int i2 = __shfl_xor(bidx, off, 32);
        if (k2 < bkey || (k2 == bkey && i2 < bidx)) { bkey = k2; bidx = i2; }
      }
      const int final_y = (bkey == 0x7fffffff) ? best_y : bidx;

      // place: owning lane updates occupancy; lane 0 writes value to own slice
      if (lane == (final_y & 31)) occ |= (1u << (final_y >> 5));
      if (lane == 0) tile[final_y * WPB + wi] = depth;
    }
  }

  __syncthreads();

  // ---- coalesced drain of the LDS tile -> out[b, 0, y, w0 + wi] ----
  const size_t obase = (size_t)b * ((size_t)H_DIM * W_DIM)
                     + (size_t)blockIdx.x * WPB;
  const int t = threadIdx.x;

#if USE_ASYNC_STORE
  // 16B groups: g in [0, H*WPB/4); y = g/(WPB/4), q = g%(WPB/4)
  constexpr int GPR = WPB / 4;                  // 16B groups per y-row (=2)
  constexpr int NG  = H_DIM * GPR;              // 1800
#pragma unroll 1
  for (int g = t; g < NG; g += TPB) {
    const int y = g / GPR;
    const int q = g - y * GPR;
    float* gp = out + obase + (size_t)y * W_DIM + q * 4;
    __builtin_amdgcn_global_store_async_from_lds_b128(
        (gv4i_p)gp, (lv4i_p)(&tile[g * 4]), 0, 0);
  }
  wait_async_cnt0();
#else
#pragma unroll 1
  for (int e = t; e < H_DIM * WPB; e += TPB) {
    const int y = e >> 3;                       // e / WPB (WPB == 8)
    const int q = e & (WPB - 1);
    out[obase + (size_t)y * W_DIM + q] = tile[e];
  }
#endif
}

extern "C" void kernel_launch(void* const* d_in, const int* in_sizes, int n_in,
                              void* d_out, int out_size, void* d_ws, size_t ws_size,
                              hipStream_t stream) {
  (void)in_sizes; (void)n_in; (void)out_size; (void)d_ws; (void)ws_size;
  const float* radar = (const float*)d_in[0];
  const float* mde   = (const float*)d_in[1];
  float* out = (float*)d_out;

  dim3 grid(W_DIM / WPB, B_DIM, 1);             // (200, 4)
  dim3 block(TPB, 1, 1);                        // 256 threads = 8 wave32
  hipLaunchKernelGGL(radar_dc_kernel, grid, block, 0, stream, radar, mde, out);
}